// GaussianHMM_29188597743662
// MI455X (gfx1250) — compile-verified
//
#include <hip/hip_runtime.h>
#include <hip/hip_bf16.h>
#include <math.h>
#include <stdint.h>

typedef __attribute__((ext_vector_type(2))) float v2f;
typedef __attribute__((ext_vector_type(4))) float v4f;
typedef __attribute__((ext_vector_type(8))) float v8f;

#define D_FEAT   1024
#define N_STATES 8
#define BATCH    256
#define TLEN     1024
#define LOG2PI   1.8378770664093453f
#define NCHUNK   (D_FEAT / 4)      // 256 K-chunks of 4
#define WCFLOATS (NCHUNK * 128)    // 32768 floats = 128 KB combined W array

__device__ __forceinline__ float sigmoidf_(float x) { return 1.0f / (1.0f + expf(-x)); }

// ---------------------------------------------------------------------------
// Prep: combined gated weight array in WMMA B-matrix layout, interleaved so the
// emission loop does ONE ds_load_b128 per K-chunk:
//   idx = c*128 + lane*4 + q ;  q in {0,1}: Wiv vgpr q ; q in {2,3}: Wm vgpr q-2
//   n = lane&15, K = (q&1) + 2*(lane>>4), d = 4*c + K.  Columns n>=8 are zero.
// Gate folded into weights: Wiv = z^2 * inv_var ; Wm = -2 * z * mu * inv_var.
// ---------------------------------------------------------------------------
__global__ void prep_w_kernel(const float* __restrict__ mu,
                              const float* __restrict__ log_var,
                              const float* __restrict__ log_alpha,
                              float* __restrict__ Wc) {
  int idx = blockIdx.x * blockDim.x + threadIdx.x;
  if (idx >= WCFLOATS) return;
  int c = idx >> 7;
  int l = (idx >> 2) & 31;
  int q = idx & 3;
  int n = l & 15;
  int k = (q & 1) + 2 * (l >> 4);
  int d = 4 * c + k;
  float val = 0.f;
  if (n < N_STATES) {
    float z = sigmoidf_(log_alpha[d]) * 1.2f - 0.1f;   // (ZETA-GAMMA)=1.2, GAMMA=-0.1
    z = fminf(fmaxf(z, 0.f), 1.f);
    float iv = expf(-log_var[n * D_FEAT + d]);
    val = (q < 2) ? (z * z * iv)
                  : (-2.f * z * mu[n * D_FEAT + d] * iv);
  }
  Wc[idx] = val;
}

// off[n] = sum_d (mu^2 * inv_var + log_var) + D*log(2*pi)
__global__ void prep_off_kernel(const float* __restrict__ mu,
                                const float* __restrict__ log_var,
                                float* __restrict__ off) {
  int n = threadIdx.x >> 5;      // 256 threads: 8 states x 32 lanes
  int lane = threadIdx.x & 31;
  float s = 0.f;
  for (int d = lane; d < D_FEAT; d += 32) {
    float lv = log_var[n * D_FEAT + d];
    float m  = mu[n * D_FEAT + d];
    s += m * m * expf(-lv) + lv;
  }
  for (int o = 16; o; o >>= 1) s += __shfl_xor(s, o);
  if (lane == 0) off[n] = s + (float)D_FEAT * LOG2PI;
}

// ---------------------------------------------------------------------------
// Emission: log_B[t, b, n] = -0.5 * (x2 - 2*xm + const + logdet + D*log(2pi))
// One wave owns one 16-row tile of flattened rows r = b*T + t.
// Weights staged through LDS in two 64 KB halves via async gather-to-LDS;
// inner loop: 1 global b64 (X) + 1 ds_load_b128 (W) + square + 2 f32 WMMAs
// into independent accumulator chains.
// ---------------------------------------------------------------------------
__global__ void emission_kernel(const float* __restrict__ X,
                                const float* __restrict__ Wc,
                                const float* __restrict__ off,
                                float* __restrict__ logB) {
  __shared__ float Wlds[WCFLOATS / 2];   // 64 KB: one K-half of the W array

  const int lane = threadIdx.x & 31;
  const int wave = threadIdx.x >> 5;
  const int tile = blockIdx.x * (blockDim.x >> 5) + wave;
  const long rbase = (long)tile * 16;

  const float* xrow = X + (rbase + (lane & 15)) * D_FEAT + 2 * (lane >> 4);

  v8f accA = {0.f, 0.f, 0.f, 0.f, 0.f, 0.f, 0.f, 0.f};
  v8f accB = {0.f, 0.f, 0.f, 0.f, 0.f, 0.f, 0.f, 0.f};

  for (int half = 0; half < 2; ++half) {
    // ---- stage 64 KB of weights: per-lane async gather into LDS ----
    const float* src = Wc + half * (WCFLOATS / 2);
    for (int i = threadIdx.x * 4; i < WCFLOATS / 2; i += 256 * 4) {
      uint32_t ldsoff = (uint32_t)(uintptr_t)&Wlds[i];   // addr[31:0] = LDS offset
      const float* gsrc = src + i;
      asm volatile("global_load_async_to_lds_b128 %0, %1, off"
                   :: "v"(ldsoff), "v"(gsrc) : "memory");
    }
    asm volatile("s_wait_asynccnt 0x0" ::: "memory");
    __syncthreads();

    // ---- compute 128 K-chunks of this half ----
    const float* xh = xrow + half * (NCHUNK / 2) * 4;
#pragma unroll 8
    for (int c = 0; c < NCHUNK / 2; ++c) {
      if ((c & 7) == 0) {
        // prefetch X ~2 KB ahead at WGP scope (scope 0: fills all cache levels)
        const float* pf = xh + c * 4 + 512;
        asm volatile("global_prefetch_b8 %0, off" :: "v"(pf));
      }
      v4f w  = *(const v4f*)&Wlds[c * 128 + lane * 4];  // ds_load_b128
      v2f xa = *(const v2f*)(xh + c * 4);               // global b64
      v2f biv = {w.x, w.y};
      v2f bm  = {w.z, w.w};
      v2f xs  = xa * xa;
      // D = A*B + C  (neg_a, A, neg_b, B, c_mod, C, reuse_a, reuse_b)
      accA = __builtin_amdgcn_wmma_f32_16x16x4_f32(false, xs, false, biv,
                                                   (short)0, accA, false, false);
      accB = __builtin_amdgcn_wmma_f32_16x16x4_f32(false, xa, false, bm,
                                                   (short)0, accB, false, false);
    }
    __syncthreads();   // protect Wlds before next half overwrites it
  }

  // C/D layout: lanes 0-15 col=lane rows M=p; lanes 16-31 col=lane-16 rows M=8+p.
  // quad[r,n] = accA + accB, both in col n (cols 8-15 zero): xor-8 adds zero col n+8.
  const float offv = off[lane & 7];
  const int b  = (int)(rbase >> 10);   // rbase / TLEN
  const int t0 = (int)(rbase & 1023);  // rbase % TLEN (tiles never straddle b)
  const int rhalf = (lane >> 4) << 3;

#pragma unroll
  for (int p = 0; p < 8; ++p) {
    float q = accA[p] + accB[p];
    q += __shfl_xor(q, 8);             // col n + col n+8 (symmetric in both lanes)
    if ((lane & 8) == 0) {
      float lb = -0.5f * (q + offv);
      int t = t0 + p + rhalf;
      logB[((long)t * BATCH + b) * N_STATES + (lane & 7)] = lb;  // [T,B,N]
    }
  }
}

// ---------------------------------------------------------------------------
// HMM forward scan: lane = (b, j).  8 lanes per batch within a wave32.
// alpha'_j = M + log( sum_i A_ij * exp(alpha_i - M) ) + logB[t,b,j]
// (exact rewrite of logsumexp_i(alpha_i + log_A_ij); A_ij >= 0 so no
//  cancellation).  1 exp + 1 log per lane per step.
// ---------------------------------------------------------------------------
__global__ void hmm_scan_kernel(const float* __restrict__ logB,   // [T,B,N]
                                const int* __restrict__ mask,     // [B,T]
                                const float* __restrict__ log_pi,
                                const float* __restrict__ log_A,  // [N,N] row i, col j
                                float* __restrict__ out) {
  const int tid  = blockIdx.x * blockDim.x + threadIdx.x;  // 0..2047
  const int b    = tid >> 3;
  const int j    = tid & 7;
  const int lane = threadIdx.x & 31;
  const int grp  = lane & ~7;
  const float NEG_INF = -INFINITY;

  float Acol[N_STATES];
#pragma unroll
  for (int i = 0; i < N_STATES; ++i) Acol[i] = expf(log_A[i * N_STATES + j]);

  float alpha = log_pi[j] + logB[(long)b * N_STATES + j];   // t = 0
  if (!mask[b * TLEN + 0]) alpha = NEG_INF;

  for (int t = 1; t < TLEN; ++t) {
    float m = alpha;
    m = fmaxf(m, __shfl_xor(m, 1));
    m = fmaxf(m, __shfl_xor(m, 2));
    m = fmaxf(m, __shfl_xor(m, 4));
    float e = expf(alpha - m);          // 0 if alpha=-inf, m finite
    float s = 0.f;
#pragma unroll
    for (int i = 0; i < N_STATES; ++i) s += Acol[i] * __shfl(e, grp + i);
    float lb = logB[((long)t * BATCH + b) * N_STATES + j];
    float anew = (m == NEG_INF) ? NEG_INF : (m + logf(s) + lb);
    if (mask[b * TLEN + t]) alpha = anew;
  }

  // final logsumexp over the 8 states
  float m = alpha;
  m = fmaxf(m, __shfl_xor(m, 1));
  m = fmaxf(m, __shfl_xor(m, 2));
  m = fmaxf(m, __shfl_xor(m, 4));
  float e = (m == NEG_INF) ? 0.f : expf(alpha - m);
  float s = e;
  s += __shfl_xor(s, 1);
  s += __shfl_xor(s, 2);
  s += __shfl_xor(s, 4);
  if (j == 0) out[b] = (m == NEG_INF) ? NEG_INF : (m + logf(s));
}

extern "C" void kernel_launch(void* const* d_in, const int* in_sizes, int n_in,
                              void* d_out, int out_size, void* d_ws, size_t ws_size,
                              hipStream_t stream) {
  (void)in_sizes; (void)n_in; (void)out_size; (void)ws_size;
  const float* X         = (const float*)d_in[0];   // [256,1024,1024]
  const int*   mask      = (const int*)  d_in[1];   // [256,1024] (bool -> int)
  const float* log_pi    = (const float*)d_in[2];   // [8]
  const float* log_A     = (const float*)d_in[3];   // [8,8]
  const float* mu        = (const float*)d_in[4];   // [8,1024]
  const float* log_var   = (const float*)d_in[5];   // [8,1024]
  const float* log_alpha = (const float*)d_in[6];   // [1024]
  float* out = (float*)d_out;                       // [256]

  // workspace: logB (8 MB) | Wc (128 KB) | off (8 f)
  float* logB = (float*)d_ws;
  float* Wc   = logB + (long)TLEN * BATCH * N_STATES;
  float* off  = Wc + WCFLOATS;

  prep_w_kernel<<<WCFLOATS / 256, 256, 0, stream>>>(mu, log_var, log_alpha, Wc);
  prep_off_kernel<<<1, 256, 0, stream>>>(mu, log_var, off);

  const int tiles = (BATCH * TLEN) / 16;            // 16384 row tiles
  emission_kernel<<<tiles / 8, 256, 0, stream>>>(X, Wc, off, logB);

  hmm_scan_kernel<<<(BATCH * N_STATES) / 256, 256, 0, stream>>>(
      logB, mask, log_pi, log_A, out);
}